// PoseProjection_18502719111434
// MI455X (gfx1250) — compile-verified
//
#include <hip/hip_runtime.h>
#include <hip/hip_bf16.h>
#include <math.h>

#define VS      0.0625f
#define INV_VS  16.0f
#define XDIM    96
#define YDIM    96
#define ZDIM    48
#define NVOX    (XDIM * YDIM * ZDIM)   // 442368
#define BATCH   8
#define CH      32

typedef float v2f __attribute__((ext_vector_type(2)));
typedef float v8f __attribute__((ext_vector_type(8)));

// ---------------------------------------------------------------------------
// Kernel 1: T_b = inv(current_pose_b) @ historical_pose_b  (general 4x4 inverse
// via adjugate, matching jnp.linalg.inv numerics closely). 8 matrices, 1 block.
// ---------------------------------------------------------------------------
__device__ __forceinline__ void invert4x4(const float* __restrict__ m, float* __restrict__ inv) {
    inv[0]  =  m[5]*m[10]*m[15] - m[5]*m[11]*m[14] - m[9]*m[6]*m[15] + m[9]*m[7]*m[14] + m[13]*m[6]*m[11] - m[13]*m[7]*m[10];
    inv[4]  = -m[4]*m[10]*m[15] + m[4]*m[11]*m[14] + m[8]*m[6]*m[15] - m[8]*m[7]*m[14] - m[12]*m[6]*m[11] + m[12]*m[7]*m[10];
    inv[8]  =  m[4]*m[9]*m[15]  - m[4]*m[11]*m[13] - m[8]*m[5]*m[15] + m[8]*m[7]*m[13] + m[12]*m[5]*m[11] - m[12]*m[7]*m[9];
    inv[12] = -m[4]*m[9]*m[14]  + m[4]*m[10]*m[13] + m[8]*m[5]*m[14] - m[8]*m[6]*m[13] - m[12]*m[5]*m[10] + m[12]*m[6]*m[9];
    inv[1]  = -m[1]*m[10]*m[15] + m[1]*m[11]*m[14] + m[9]*m[2]*m[15] - m[9]*m[3]*m[14] - m[13]*m[2]*m[11] + m[13]*m[3]*m[10];
    inv[5]  =  m[0]*m[10]*m[15] - m[0]*m[11]*m[14] - m[8]*m[2]*m[15] + m[8]*m[3]*m[14] + m[12]*m[2]*m[11] - m[12]*m[3]*m[10];
    inv[9]  = -m[0]*m[9]*m[15]  + m[0]*m[11]*m[13] + m[8]*m[1]*m[15] - m[8]*m[3]*m[13] - m[12]*m[1]*m[11] + m[12]*m[3]*m[9];
    inv[13] =  m[0]*m[9]*m[14]  - m[0]*m[10]*m[13] - m[8]*m[1]*m[14] + m[8]*m[2]*m[13] + m[12]*m[1]*m[10] - m[12]*m[2]*m[9];
    inv[2]  =  m[1]*m[6]*m[15]  - m[1]*m[7]*m[14]  - m[5]*m[2]*m[15] + m[5]*m[3]*m[14] + m[13]*m[2]*m[7]  - m[13]*m[3]*m[6];
    inv[6]  = -m[0]*m[6]*m[15]  + m[0]*m[7]*m[14]  + m[4]*m[2]*m[15] - m[4]*m[3]*m[14] - m[12]*m[2]*m[7]  + m[12]*m[3]*m[6];
    inv[10] =  m[0]*m[5]*m[15]  - m[0]*m[7]*m[13]  - m[4]*m[1]*m[15] + m[4]*m[3]*m[13] + m[12]*m[1]*m[7]  - m[12]*m[3]*m[5];
    inv[14] = -m[0]*m[5]*m[14]  + m[0]*m[6]*m[13]  + m[4]*m[1]*m[14] - m[4]*m[2]*m[13] - m[12]*m[1]*m[6]  + m[12]*m[2]*m[5];
    inv[3]  = -m[1]*m[6]*m[11]  + m[1]*m[7]*m[10]  + m[5]*m[2]*m[11] - m[5]*m[3]*m[10] - m[9]*m[2]*m[7]   + m[9]*m[3]*m[6];
    inv[7]  =  m[0]*m[6]*m[11]  - m[0]*m[7]*m[10]  - m[4]*m[2]*m[11] + m[4]*m[3]*m[10] + m[8]*m[2]*m[7]   - m[8]*m[3]*m[6];
    inv[11] = -m[0]*m[5]*m[11]  + m[0]*m[7]*m[9]   + m[4]*m[1]*m[11] - m[4]*m[3]*m[9]  - m[8]*m[1]*m[7]   + m[8]*m[3]*m[5];
    inv[15] =  m[0]*m[5]*m[10]  - m[0]*m[6]*m[9]   - m[4]*m[1]*m[10] + m[4]*m[2]*m[9]  + m[8]*m[1]*m[6]   - m[8]*m[2]*m[5];
    float det = m[0]*inv[0] + m[1]*inv[4] + m[2]*inv[8] + m[3]*inv[12];
    float r = 1.0f / det;
    #pragma unroll
    for (int i = 0; i < 16; ++i) inv[i] *= r;
}

__global__ void pose_compose_kernel(const float* __restrict__ hist,
                                    const float* __restrict__ cur,
                                    float* __restrict__ T) {
    int b = threadIdx.x;
    if (b >= BATCH) return;
    float inv[16];
    invert4x4(cur + b * 16, inv);
    const float* h = hist + b * 16;
    #pragma unroll
    for (int i = 0; i < 4; ++i) {
        #pragma unroll
        for (int j = 0; j < 4; ++j) {
            float s = 0.0f;
            #pragma unroll
            for (int k = 0; k < 4; ++k) s += inv[i * 4 + k] * h[k * 4 + j];
            T[b * 16 + i * 4 + j] = s;
        }
    }
}

// ---------------------------------------------------------------------------
// Kernel 2: one lane per output voxel. Coordinate transform done with
// V_WMMA_F32_16X16X4_F32:  A = transform rows (16x4, rows 0..3 valid),
// B columns = 16 voxels' homogeneous coords (4x16). D row m, col n =
// component m of transformed voxel n  ->  lane n holds x/y/z in D[0..2].
// Two WMMAs per wave cover 32 voxels; shuffles redistribute to the owner lane.
// ---------------------------------------------------------------------------
__global__ __launch_bounds__(256) void project_kernel(
    const float* __restrict__ feat,   // [B,32,48,96,96]
    const float* __restrict__ sdf,    // [B, 1,48,96,96]
    const float* __restrict__ occ,    // [B, 1,48,96,96]
    const float* __restrict__ T,      // [B,4,4]
    float* __restrict__ out_feat,
    float* __restrict__ out_sdf,
    float* __restrict__ out_occ,
    float* __restrict__ out_grid) {

    const int tid  = blockIdx.x * blockDim.x + threadIdx.x;   // grid exactly covers B*NVOX
    const int vox  = tid % NVOX;
    const int b    = tid / NVOX;
    const int w    = vox % XDIM;
    const int h    = (vox / XDIM) % YDIM;
    const int d    = vox / (XDIM * YDIM);
    const int lane = threadIdx.x & 31;
    const int m    = lane & 15;                // row (A) / column (B) index
    const bool lo  = lane < 16;
    const int wbase = w - lane;                // wave-uniform (32 | 96)

    // ---- A: transform rows. lane<16 -> (K=0,K=1); lane>=16 -> (K=2,K=3) ----
    const float* Tb = T + b * 16;
    float a0 = (m < 4) ? Tb[m * 4 + (lo ? 0 : 2)] : 0.0f;
    float a1 = (m < 4) ? Tb[m * 4 + (lo ? 1 : 3)] : 0.0f;
    v2f A; A[0] = a0; A[1] = a1;

    // ---- B: homogeneous voxel coords, column n = voxel (wbase [+16] + n) ----
    const float py = (float)h * VS;
    const float pz = (float)d * VS;
    float px1 = (float)(wbase + m) * VS;
    float px2 = (float)(wbase + 16 + m) * VS;
    v2f B1; B1[0] = lo ? px1 : pz;  B1[1] = lo ? py : 1.0f;
    v2f B2; B2[0] = lo ? px2 : pz;  B2[1] = lo ? py : 1.0f;

    v8f c = {};
    v8f D1 = __builtin_amdgcn_wmma_f32_16x16x4_f32(false, A, false, B1, (short)0, c, false, false);
    v8f D2 = __builtin_amdgcn_wmma_f32_16x16x4_f32(false, A, false, B2, (short)0, c, false, false);

    // ---- redistribute: lane l's voxel is column (l&15) of D1 (l<16) / D2 ----
    float x1 = __shfl(D1[0], m, 32), x2 = __shfl(D2[0], m, 32);
    float y1 = __shfl(D1[1], m, 32), y2 = __shfl(D2[1], m, 32);
    float z1 = __shfl(D1[2], m, 32), z2 = __shfl(D2[2], m, 32);
    float ix = (lo ? x1 : x2) * INV_VS;   // sampling coords: (grid+1)*0.5*(dim-1) == hist/VS
    float iy = (lo ? y1 : y2) * INV_VS;
    float iz = (lo ? z1 : z2) * INV_VS;

    // ---- grid output [B,D,H,W,3] ----
    {
        const float gx = 2.0f * ix * (1.0f / 95.0f) - 1.0f;
        const float gy = 2.0f * iy * (1.0f / 95.0f) - 1.0f;
        const float gz = 2.0f * iz * (1.0f / 47.0f) - 1.0f;
        float* g = out_grid + (size_t)tid * 3;
        g[0] = gx; g[1] = gy; g[2] = gz;
    }

    // ---- trilinear setup: 8 corner offsets + masked weights ----
    const float fx0 = floorf(ix), fy0 = floorf(iy), fz0 = floorf(iz);
    const float fx = ix - fx0, fy = iy - fy0, fz = iz - fz0;
    const int x0 = (int)fx0, y0 = (int)fy0, z0 = (int)fz0;

    float wgt[8];
    int   off[8];
    #pragma unroll
    for (int k = 0; k < 8; ++k) {
        const int dx = k & 1, dy = (k >> 1) & 1, dz = k >> 2;
        const int xi = x0 + dx, yi = y0 + dy, zi = z0 + dz;
        const bool valid = (xi >= 0) & (xi <= XDIM - 1) &
                           (yi >= 0) & (yi <= YDIM - 1) &
                           (zi >= 0) & (zi <= ZDIM - 1);
        const int xc = min(max(xi, 0), XDIM - 1);
        const int yc = min(max(yi, 0), YDIM - 1);
        const int zc = min(max(zi, 0), ZDIM - 1);
        const float wk = (dx ? fx : 1.0f - fx) * (dy ? fy : 1.0f - fy) * (dz ? fz : 1.0f - fz);
        wgt[k] = valid ? wk : 0.0f;
        off[k] = (zc * YDIM + yc) * XDIM + xc;
    }

    // ---- features: 32 channels, coalesced per-channel writes ----
    const float* fb  = feat + (size_t)b * CH * NVOX;
    float*       ofb = out_feat + (size_t)b * CH * NVOX + vox;
    #pragma unroll 8
    for (int ch = 0; ch < CH; ++ch) {
        const float* fc = fb + (size_t)ch * NVOX;
        float acc = 0.0f;
        #pragma unroll
        for (int k = 0; k < 8; ++k) acc += wgt[k] * fc[off[k]];
        ofb[(size_t)ch * NVOX] = acc;
    }

    // ---- sdf / occupancy (C = 1) ----
    {
        const float* sb = sdf + (size_t)b * NVOX;
        const float* ob = occ + (size_t)b * NVOX;
        float accs = 0.0f, acco = 0.0f;
        #pragma unroll
        for (int k = 0; k < 8; ++k) {
            accs += wgt[k] * sb[off[k]];
            acco += wgt[k] * ob[off[k]];
        }
        out_sdf[(size_t)b * NVOX + vox] = accs;
        out_occ[(size_t)b * NVOX + vox] = acco;
    }
}

// ---------------------------------------------------------------------------
extern "C" void kernel_launch(void* const* d_in, const int* in_sizes, int n_in,
                              void* d_out, int out_size, void* d_ws, size_t ws_size,
                              hipStream_t stream) {
    const float* feat = (const float*)d_in[0];
    const float* sdf  = (const float*)d_in[1];
    const float* occ  = (const float*)d_in[2];
    const float* hist = (const float*)d_in[3];
    const float* cur  = (const float*)d_in[4];

    float* T = (float*)d_ws;                       // 8 * 16 floats = 512 B

    float* out      = (float*)d_out;
    float* out_feat = out;                                          // [8,32,48,96,96]
    float* out_sdf  = out_feat + (size_t)BATCH * CH * NVOX;         // [8,1,48,96,96]
    float* out_occ  = out_sdf  + (size_t)BATCH * NVOX;              // [8,1,48,96,96]
    float* out_grid = out_occ  + (size_t)BATCH * NVOX;              // [8,48,96,96,3]

    pose_compose_kernel<<<1, 32, 0, stream>>>(hist, cur, T);

    const int total  = BATCH * NVOX;               // 3,538,944 (divisible by 256)
    const int block  = 256;
    project_kernel<<<total / block, block, 0, stream>>>(
        feat, sdf, occ, T, out_feat, out_sdf, out_occ, out_grid);
}